// MultiEncoding__79680233275629
// MI455X (gfx1250) — compile-verified
//
#include <hip/hip_runtime.h>

// CDNA5 (gfx1250) fused quanvolution + MLP kernel.
// One image per lane; 196-step quantum-window scan fully in registers;
// fc1 accumulated incrementally with V_WMMA_F32_16X16X4_F32 (M=16,K=4 f32).

typedef float v2f __attribute__((ext_vector_type(2)));
typedef float v8f __attribute__((ext_vector_type(8)));

#define H1S 21  // LDS row stride (stride 21 -> 32 distinct banks, conflict-free)

__global__ __launch_bounds__(256) void quanv_fused_kernel(
    const float* __restrict__ x,      // (B,1,28,28)
    const float* __restrict__ crz_p,  // scalar
    const float* __restrict__ ry_p,   // scalar
    const float* __restrict__ fc1w,   // (20,784) row-major
    const float* __restrict__ fc1b,   // (20,)
    const float* __restrict__ fc2w,   // (2,20)  row-major
    const float* __restrict__ fc2b,   // (2,)
    float* __restrict__ out,          // (B,2)
    int B)
{
    __shared__ float lds_h1[8 * 32 * H1S];

    const int tid   = threadIdx.x;
    const int lane  = tid & 31;
    const int wv    = tid >> 5;
    const int gid   = blockIdx.x * 256 + tid;
    const bool valid = (gid < B);
    const int img   = valid ? gid : (B - 1);   // clamp: keep EXEC all-ones for WMMA
    const float* __restrict__ ximg = x + (size_t)img * 784;

    // ---- uniform gate constants -------------------------------------------
    const float crz = crz_p[0];
    const float ryt = ry_p[0];
    float rys, ryc;
    __sincosf(0.5f * ryt, &rys, &ryc);

    // CRZ diagonal: exp(+0.5i * theta * sign[b]); sign table from reference
    const int csign[16] = {0,-1,-1,0,-1,-2,0,1,-1,0,-2,1,0,1,1,4};
    float cdr[16], cdi[16];
#pragma unroll
    for (int b = 0; b < 16; ++b) {
        float sv, cv;
        __sincosf(0.5f * crz * (float)csign[b], &sv, &cv);
        cdr[b] = cv; cdi[b] = sv;
    }

    // ---- quantum state: 16 complex amps in registers ----------------------
    float sr[16], si[16];
#pragma unroll
    for (int b = 0; b < 16; ++b) { sr[b] = 0.f; si[b] = 0.f; }
    sr[0] = 1.f;

    // ---- WMMA accumulators: h1 (32 images x [16 | 16-padded-20]) per wave -
    v8f acc00 = {}, acc01 = {}, acc10 = {}, acc11 = {};

    const float inv_sqrt2 = 0.70710678118654752440f;
    const int   n16   = lane & 15;
    const int   kh    = (lane & 16) ? 2 : 0;          // K half for A/B fragments
    const float t1m   = (n16 < 4) ? 1.f : 0.f;        // tile-1 valid-col mask
    const int   colt1 = 16 + ((n16 < 4) ? n16 : 3);   // clamped col for tile 1

    for (int s = 0; s < 196; ++s) {
        const int i0 = 2 * (s / 14);
        const int j0 = 2 * (s % 14);
        const int h  = (i0 == 26) ? 2 : 4;
        const int w  = (j0 == 26) ? 2 : 4;
        const int hw = h * w;
        const int wsh   = (w == 4) ? 2 : 1;
        const int wmask = w - 1;

        // window features in the reference's *packed* order (then zero-pad)
        float f[16];
#pragma unroll
        for (int k = 0; k < 16; ++k) {
            const int kk = (k < hw) ? k : 0;          // clamp -> always in-bounds
            const int r = kk >> wsh, c = kk & wmask;
            float v = ximg[(i0 + r) * 28 + (j0 + c)];
            f[k] = (k < hw) ? v : 0.f;
        }
        if (s + 1 < 196) {  // gfx1250 global_prefetch_b8 of next window row
            const int ni = 2 * ((s + 1) / 14), nj = 2 * ((s + 1) % 14);
            __builtin_prefetch(ximg + ni * 28 + nj, 0, 0);
        }

        // ---- per-qubit gate build (U = M_n ... M_1 @ H) + apply -----------
#pragma unroll
        for (int wq = 0; wq < 4; ++wq) {
            float u00r = inv_sqrt2, u00i = 0.f, u01r =  inv_sqrt2, u01i = 0.f;
            float u10r = inv_sqrt2, u10i = 0.f, u11r = -inv_sqrt2, u11i = 0.f;
            const int ng = (wq < 3) ? 5 : 1;
#pragma unroll
            for (int g = 0; g < ng; ++g) {
                float sa, ca;
                __sincosf(0.5f * f[wq * 5 + g], &sa, &ca);
                if ((g & 1) == 0) {
                    // RZ: row0 *= (ca - i sa); row1 *= (ca + i sa)
                    float r0, i0v;
                    r0 = u00r*ca + u00i*sa; i0v = u00i*ca - u00r*sa; u00r=r0; u00i=i0v;
                    r0 = u01r*ca + u01i*sa; i0v = u01i*ca - u01r*sa; u01r=r0; u01i=i0v;
                    r0 = u10r*ca - u10i*sa; i0v = u10i*ca + u10r*sa; u10r=r0; u10i=i0v;
                    r0 = u11r*ca - u11i*sa; i0v = u11i*ca + u11r*sa; u11r=r0; u11i=i0v;
                } else {
                    // RY: row0' = c*row0 - s*row1; row1' = s*row0 + c*row1
                    float n00r = ca*u00r - sa*u10r, n00i = ca*u00i - sa*u10i;
                    float n01r = ca*u01r - sa*u11r, n01i = ca*u01i - sa*u11i;
                    float n10r = sa*u00r + ca*u10r, n10i = sa*u00i + ca*u10i;
                    float n11r = sa*u01r + ca*u11r, n11i = sa*u01i + ca*u11i;
                    u00r=n00r; u00i=n00i; u01r=n01r; u01i=n01i;
                    u10r=n10r; u10i=n10i; u11r=n11r; u11i=n11i;
                }
            }
            const int R = 8 >> wq;                    // qubit bit stride
#pragma unroll
            for (int idx = 0; idx < 16; ++idx) {
                if (idx & R) continue;
                float ar = sr[idx],     ai = si[idx];
                float br = sr[idx + R], bi = si[idx + R];
                sr[idx]     = u00r*ar - u00i*ai + u01r*br - u01i*bi;
                si[idx]     = u00r*ai + u00i*ar + u01r*bi + u01i*br;
                sr[idx + R] = u10r*ar - u10i*ai + u11r*br - u11i*bi;
                si[idx + R] = u10r*ai + u10i*ar + u11r*bi + u11i*br;
            }
        }

        // ---- CRZ diagonal --------------------------------------------------
#pragma unroll
        for (int b = 0; b < 16; ++b) {
            float rr = sr[b]*cdr[b] - si[b]*cdi[b];
            float ii = sr[b]*cdi[b] + si[b]*cdr[b];
            sr[b] = rr; si[b] = ii;
        }
        // ---- fixed RY layer ------------------------------------------------
#pragma unroll
        for (int wq = 0; wq < 4; ++wq) {
            const int R = 8 >> wq;
#pragma unroll
            for (int idx = 0; idx < 16; ++idx) {
                if (idx & R) continue;
                float ar = sr[idx],     ai = si[idx];
                float br = sr[idx + R], bi = si[idx + R];
                sr[idx]     = ryc*ar - rys*br;  si[idx]     = ryc*ai - rys*bi;
                sr[idx + R] = rys*ar + ryc*br;  si[idx + R] = rys*ai + ryc*bi;
            }
        }

        // ---- Z expectations (4 features / image) --------------------------
        float f0 = 0.f, f1 = 0.f, f2 = 0.f, f3 = 0.f;
#pragma unroll
        for (int b = 0; b < 16; ++b) {
            float p = sr[b]*sr[b] + si[b]*si[b];
            f0 += (b & 8) ? -p : p;
            f1 += (b & 4) ? -p : p;
            f2 += (b & 2) ? -p : p;
            f3 += (b & 1) ? -p : p;
        }

        // ---- incremental fc1 via V_WMMA_F32_16X16X4_F32 -------------------
        // A (16x4): vgpr0 = K0|K2 , vgpr1 = K1|K3 across lane halves
        float t2 = __shfl(f2, n16, 32);
        float t3 = __shfl(f3, n16, 32);
        float a0 = (lane < 16) ? f0 : t2;       // images 0..15
        float a1 = (lane < 16) ? f1 : t3;
        float u0 = __shfl(f0, n16 | 16, 32);
        float u1 = __shfl(f1, n16 | 16, 32);
        float g0 = (lane < 16) ? u0 : f2;       // images 16..31
        float g1 = (lane < 16) ? u1 : f3;
        v2f A0; A0.x = a0; A0.y = a1;
        v2f A1; A1.x = g0; A1.y = g1;
        // B (4x16): B[k][col] = fc1w[col*784 + 4s + k]  (broadcast, L2-resident)
        const int s4 = 4 * s;
        const float* wp0 = fc1w + n16 * 784 + s4 + kh;
        const float* wp1 = fc1w + colt1 * 784 + s4 + kh;
        v2f B0; B0.x = wp0[0];       B0.y = wp0[1];
        v2f B1; B1.x = wp1[0] * t1m; B1.y = wp1[1] * t1m;

        acc00 = __builtin_amdgcn_wmma_f32_16x16x4_f32(false, A0, false, B0, (short)0, acc00, false, false);
        acc01 = __builtin_amdgcn_wmma_f32_16x16x4_f32(false, A0, false, B1, (short)0, acc01, false, false);
        acc10 = __builtin_amdgcn_wmma_f32_16x16x4_f32(false, A1, false, B0, (short)0, acc10, false, false);
        acc11 = __builtin_amdgcn_wmma_f32_16x16x4_f32(false, A1, false, B1, (short)0, acc11, false, false);
    }

    // ---- epilogue: D tiles -> LDS transpose -> bias/leaky-ReLU/fc2 --------
    float* hl = &lds_h1[wv * 32 * H1S];
    const int rb = (lane & 16) ? 8 : 0;
#pragma unroll
    for (int r = 0; r < 8; ++r) {
        hl[(rb + r) * H1S + n16]            = acc00[r];
        hl[(16 + rb + r) * H1S + n16]       = acc10[r];
        if (n16 < 4) {
            hl[(rb + r) * H1S + 16 + n16]      = acc01[r];
            hl[(16 + rb + r) * H1S + 16 + n16] = acc11[r];
        }
    }
    __syncthreads();

    float h1[20];
#pragma unroll
    for (int n = 0; n < 20; ++n) {
        float v = hl[lane * H1S + n] + fc1b[n];
        h1[n] = (v > 0.f) ? v : 0.1f * v;
    }
    float o0 = fc2b[0], o1 = fc2b[1];
#pragma unroll
    for (int n = 0; n < 20; ++n) {
        o0 += h1[n] * fc2w[n];
        o1 += h1[n] * fc2w[20 + n];
    }
    if (valid) {
        float2 o; o.x = o0; o.y = o1;
        *(float2*)(out + (size_t)img * 2) = o;
    }
}

extern "C" void kernel_launch(void* const* d_in, const int* in_sizes, int n_in,
                              void* d_out, int out_size, void* d_ws, size_t ws_size,
                              hipStream_t stream) {
    const float* x    = (const float*)d_in[0];
    const float* crz  = (const float*)d_in[1];
    const float* ryt  = (const float*)d_in[2];
    const float* fc1w = (const float*)d_in[3];
    const float* fc1b = (const float*)d_in[4];
    const float* fc2w = (const float*)d_in[5];
    const float* fc2b = (const float*)d_in[6];
    float* outp = (float*)d_out;

    const int B = in_sizes[0] / 784;
    const int blocks = (B + 255) / 256;
    quanv_fused_kernel<<<blocks, 256, 0, stream>>>(x, crz, ryt, fc1w, fc1b,
                                                   fc2w, fc2b, outp, B);
}